// SDFGrid_42219528519911
// MI455X (gfx1250) — compile-verified
//
#include <hip/hip_runtime.h>

// Grid volume dimensions (reference RES = (128,128,128)), channel 0 only.
#define GD 128
#define GH 128
#define GW 128

typedef float v4f __attribute__((ext_vector_type(4)));
typedef float v2f __attribute__((ext_vector_type(2)));

// Load 8 bytes (two adjacent floats) from a 4-byte-aligned address.
// memcpy form avoids alignment UB; gfx1250 supports unaligned global b64,
// so this lowers to a single global_load_b64 gather.
__device__ __forceinline__ v2f load_pair(const float* __restrict__ p) {
    v2f q;
    __builtin_memcpy(&q, p, sizeof(q));
    return q;
}

// Trilinear sample matching jax reference (align_corners=True semantics) for
// in-range coords. x -> W axis, y -> H axis, z -> D axis; vol[z,y,x].
__device__ __forceinline__ float tri_one(const float* __restrict__ vol,
                                         float px, float py, float pz,
                                         float mnx, float mny, float mnz,
                                         float sx, float sy, float sz)
{
    // ix = (p - min)/(max - min) * (W-1)  ==  ((nc+1)*0.5*(W-1)) of the reference
    float ix = (px - mnx) * sx;
    float iy = (py - mny) * sy;
    float iz = (pz - mnz) * sz;
    // Defensive clamp of continuous coords (no-op for in-range inputs).
    ix = __builtin_fminf(__builtin_fmaxf(ix, 0.0f), (float)(GW - 1));
    iy = __builtin_fminf(__builtin_fmaxf(iy, 0.0f), (float)(GH - 1));
    iz = __builtin_fminf(__builtin_fmaxf(iz, 0.0f), (float)(GD - 1));

    // x0 in [0, W-2] so x1 = x0+1 is always in range; wx = ix - x0 matches the
    // reference floor/clamp construction (incl. the ix == W-1 limit case).
    int x0 = min((int)ix, GW - 2);
    int y0 = min((int)iy, GH - 2);
    int z0 = min((int)iz, GD - 2);
    float wx = ix - (float)x0;
    float wy = iy - (float)y0;
    float wz = iz - (float)z0;

    int base = (z0 * GH + y0) * GW + x0;
    // Four b64 gathers instead of eight b32: x-pairs are contiguous.
    v2f q00 = load_pair(vol + base);                 // (x0..x1, y0, z0)
    v2f q01 = load_pair(vol + base + GW);            // (x0..x1, y1, z0)
    v2f q10 = load_pair(vol + base + GH * GW);       // (x0..x1, y0, z1)
    v2f q11 = load_pair(vol + base + GH * GW + GW);  // (x0..x1, y1, z1)

    // Mirror the reference arithmetic form a*(1-w) + b*w.
    float omx = 1.0f - wx;
    float c00 = q00.x * omx + q00.y * wx;
    float c01 = q01.x * omx + q01.y * wx;
    float c10 = q10.x * omx + q10.y * wx;
    float c11 = q11.x * omx + q11.y * wx;
    float c0  = c00 * (1.0f - wy) + c01 * wy;
    float c1  = c10 * (1.0f - wy) + c11 * wy;
    return c0 * (1.0f - wz) + c1 * wz;
}

__global__ __launch_bounds__(256) void sdf_tri_kernel(
    const float* __restrict__ points,   // [N,3]
    const float* __restrict__ vol,      // 128^3, channel 0 of grid
    const float* __restrict__ minb,     // [3]
    const float* __restrict__ maxb,     // [3]
    float* __restrict__ out,            // [N]
    int n)
{
    const int nquads = (n + 3) >> 2;
    int t = blockIdx.x * blockDim.x + threadIdx.x;
    if (t >= nquads) return;

    // Uniform bounds -> scalar loads. scale = (dim-1) * rcp(max-min).
    // v_rcp_f32 is exact for power-of-two ranges (here max-min == 1.0) and
    // 1-ulp otherwise -- avoids the full IEEE v_div_scale/fixup chain.
    float mnx = minb[0], mny = minb[1], mnz = minb[2];
    float sx = (float)(GW - 1) * __builtin_amdgcn_rcpf(maxb[0] - mnx);
    float sy = (float)(GH - 1) * __builtin_amdgcn_rcpf(maxb[1] - mny);
    float sz = (float)(GD - 1) * __builtin_amdgcn_rcpf(maxb[2] - mnz);

    int i0 = t << 2;
    if (i0 + 4 <= n) {
        // Fast path: 4 points = 12 floats = three coalesced b128 NT loads.
        const v4f* p4 = (const v4f*)points;
        v4f A = __builtin_nontemporal_load(p4 + 3 * t + 0);
        v4f B = __builtin_nontemporal_load(p4 + 3 * t + 1);
        v4f C = __builtin_nontemporal_load(p4 + 3 * t + 2);

        v4f r;
        r.x = tri_one(vol, A.x, A.y, A.z, mnx, mny, mnz, sx, sy, sz);
        r.y = tri_one(vol, A.w, B.x, B.y, mnx, mny, mnz, sx, sy, sz);
        r.z = tri_one(vol, B.z, B.w, C.x, mnx, mny, mnz, sx, sy, sz);
        r.w = tri_one(vol, C.y, C.z, C.w, mnx, mny, mnz, sx, sy, sz);

        __builtin_nontemporal_store(r, (v4f*)out + t);
    } else {
        // Tail (n not a multiple of 4).
        for (int i = i0; i < n; ++i) {
            float px = points[3 * i + 0];
            float py = points[3 * i + 1];
            float pz = points[3 * i + 2];
            out[i] = tri_one(vol, px, py, pz, mnx, mny, mnz, sx, sy, sz);
        }
    }
}

extern "C" void kernel_launch(void* const* d_in, const int* in_sizes, int n_in,
                              void* d_out, int out_size, void* d_ws, size_t ws_size,
                              hipStream_t stream) {
    // setup_inputs() order: points [N,3], grid [1,28,128,128,128], min_bound [3], max_bound [3]
    const float* points = (const float*)d_in[0];
    const float* grid   = (const float*)d_in[1];   // channel 0 == first 128^3 floats
    const float* minb   = (const float*)d_in[2];
    const float* maxb   = (const float*)d_in[3];
    float* out = (float*)d_out;

    int n = in_sizes[0] / 3;          // 8,000,000
    int nquads = (n + 3) / 4;         // 2,000,000 threads, 4 points each
    const int block = 256;            // 8 wave32 waves per block
    int gridDim = (nquads + block - 1) / block;

    sdf_tri_kernel<<<gridDim, block, 0, stream>>>(points, grid, minb, maxb, out, n);
}